// GNN_7232724927094
// MI455X (gfx1250) — compile-verified
//
#include <hip/hip_runtime.h>
#include <hip/hip_bf16.h>

typedef float v2f __attribute__((ext_vector_type(2)));
typedef float v8f __attribute__((ext_vector_type(8)));

#define Dg 300
#define Hg 600
#define Lg 5
#define NGg 512
#define BN_EPS_C 1e-5f
#define HP 640   // padded Hg (multiple of 64)
#define DP 320   // padded Dg (multiple of 64)

// h[n,:] = sum_f atom_emb[x[n,f] + f*64, :]   (one wave per node)
__global__ void atom_encode_k(float* __restrict__ h, const int* __restrict__ x,
                              const float* __restrict__ aemb, int N) {
  int node = (int)((blockIdx.x * blockDim.x + threadIdx.x) >> 5);
  int lane = threadIdx.x & 31;
  if (node >= N) return;
  const int* xr = x + node * 9;
  int rows[9];
#pragma unroll
  for (int f = 0; f < 9; ++f) rows[f] = xr[f] + f * 64;
  float* hr = h + (size_t)node * Dg;
  for (int d = lane; d < Dg; d += 32) {
    float s = 0.f;
#pragma unroll
    for (int f = 0; f < 9; ++f) s += aemb[(size_t)rows[f] * Dg + d];
    hr[d] = s;
  }
}

// tab[c,:] = bemb[i0,:] + bemb[8+i1,:] + bemb[16+i2,:], c = i0 + 8*i1 + 64*i2
__global__ void bond_combine_k(float* __restrict__ tab, const float* __restrict__ bemb) {
  int c = blockIdx.x;
  int i0 = c & 7, i1 = (c >> 3) & 7, i2 = (c >> 6) & 7;
  const float* r0 = bemb + (size_t)i0 * Dg;
  const float* r1 = bemb + (size_t)(8 + i1) * Dg;
  const float* r2 = bemb + (size_t)(16 + i2) * Dg;
  float* tr = tab + (size_t)c * Dg;
  for (int d = threadIdx.x; d < Dg; d += blockDim.x) tr[d] = r0[d] + r1[d] + r2[d];
}

// z = (1 + eps[l]) * h
__global__ void z_init_k(float* __restrict__ z, const float* __restrict__ h,
                         const float* __restrict__ eps, int l, int total) {
  float s = 1.0f + eps[l];
  for (int i = blockIdx.x * blockDim.x + threadIdx.x; i < total;
       i += gridDim.x * blockDim.x)
    z[i] = s * h[i];
}

// z[dst,:] += relu(h[src,:] + tab[code,:])   (one wave per edge)
__global__ void msg_scatter_k(float* __restrict__ z, const float* __restrict__ h,
                              const float* __restrict__ tab,
                              const int* __restrict__ src, const int* __restrict__ dst,
                              const int* __restrict__ eattr, int E) {
  int e = (int)((blockIdx.x * blockDim.x + threadIdx.x) >> 5);
  int lane = threadIdx.x & 31;
  if (e >= E) return;
  int s = src[e], t = dst[e];
  const int* ea = eattr + (size_t)e * 3;
  int code = ea[0] + (ea[1] << 3) + (ea[2] << 6);
  const float* hr = h + (size_t)s * Dg;
  const float* tr = tab + (size_t)code * Dg;
  float* zr = z + (size_t)t * Dg;
  for (int d = lane; d < Dg; d += 32) {
    float v = hr[d] + tr[d];
    v = v > 0.f ? v : 0.f;
    atomicAdd(zr + d, v);
  }
}

// Wp[K x Ncp] = W[K x Nc] zero-padded in columns
__global__ void pad_w_k(float* __restrict__ Wp, const float* __restrict__ W,
                        int K, int Nc, int Ncp) {
  int total = K * Ncp;
  for (int i = blockIdx.x * blockDim.x + threadIdx.x; i < total;
       i += gridDim.x * blockDim.x) {
    int r = i / Ncp, c = i - r * Ncp;
    Wp[i] = (c < Nc) ? W[(size_t)r * Nc + c] : 0.f;
  }
}

// C[M,Nc] = A[M,K] @ Bp[K,Ncp] + bias, via V_WMMA_F32_16X16X4_F32.
// Bp column-padded (Ncp multiple of 64) -> branch-free inner loop.
// One wave -> 16x64 output tile (4 accumulators). M%16==0, K%4==0.
// Software-pipelined: iteration k+1 fragments are loaded before the WMMAs
// consuming iteration k, so the wait before the WMMA quad is partial.
template <int K>
__global__ void __launch_bounds__(256)
gemm_bias_k(float* __restrict__ C, const float* __restrict__ A,
            const float* __restrict__ Bp, const float* __restrict__ bias,
            int M, int Nc, int Ncp) {
  int lane = threadIdx.x & 31;
  int wave = threadIdx.x >> 5;
  int rowTile = blockIdx.y * 8 + wave;
  if (rowTile * 16 >= M) return;          // wave-uniform: EXEC stays all-ones
  int m0 = rowTile * 16;
  int n0 = blockIdx.x * 64;

  v8f acc0 = {}, acc1 = {}, acc2 = {}, acc3 = {};
  int ksub = (lane >> 4) * 2;             // 0 or 2
  int nloc = lane & 15;
  const float* Arow = A + (size_t)(m0 + (lane & 15)) * K + ksub;
  const float* Bcol = Bp + (size_t)ksub * Ncp + n0 + nloc;

  // Prologue: fragments for k0 = 0
  v2f a = *(const v2f*)(Arow);
  v2f b0 = {Bcol[0],  Bcol[Ncp]};
  v2f b1 = {Bcol[16], Bcol[Ncp + 16]};
  v2f b2 = {Bcol[32], Bcol[Ncp + 32]};
  v2f b3 = {Bcol[48], Bcol[Ncp + 48]};

  constexpr int NITER = K / 4;
#pragma unroll 5
  for (int it = 0; it < NITER - 1; ++it) {
    int kn = (it + 1) * 4;
    // Prefetch next iteration's fragments (independent of the WMMAs below)
    v2f an = *(const v2f*)(Arow + kn);
    const float* Bn = Bcol + (size_t)kn * Ncp;
    v2f c0 = {Bn[0],  Bn[Ncp]};
    v2f c1 = {Bn[16], Bn[Ncp + 16]};
    v2f c2 = {Bn[32], Bn[Ncp + 32]};
    v2f c3 = {Bn[48], Bn[Ncp + 48]};
    // Consume current fragments
    acc0 = __builtin_amdgcn_wmma_f32_16x16x4_f32(false, a, false, b0, (short)0, acc0, false, false);
    acc1 = __builtin_amdgcn_wmma_f32_16x16x4_f32(false, a, false, b1, (short)0, acc1, false, false);
    acc2 = __builtin_amdgcn_wmma_f32_16x16x4_f32(false, a, false, b2, (short)0, acc2, false, false);
    acc3 = __builtin_amdgcn_wmma_f32_16x16x4_f32(false, a, false, b3, (short)0, acc3, false, false);
    a = an; b0 = c0; b1 = c1; b2 = c2; b3 = c3;
  }
  // Epilogue: last iteration
  acc0 = __builtin_amdgcn_wmma_f32_16x16x4_f32(false, a, false, b0, (short)0, acc0, false, false);
  acc1 = __builtin_amdgcn_wmma_f32_16x16x4_f32(false, a, false, b1, (short)0, acc1, false, false);
  acc2 = __builtin_amdgcn_wmma_f32_16x16x4_f32(false, a, false, b2, (short)0, acc2, false, false);
  acc3 = __builtin_amdgcn_wmma_f32_16x16x4_f32(false, a, false, b3, (short)0, acc3, false, false);

  int mrBase = m0 + (lane >> 4) * 8;      // C/D layout: lanes>=16 hold M=r+8
#pragma unroll
  for (int r = 0; r < 8; ++r) {
    float* Crow = C + (size_t)(mrBase + r) * Nc;
    int n = n0 + nloc;
    if (n < Nc) Crow[n] = acc0[r] + bias[n];
    n += 16; if (n < Nc) Crow[n] = acc1[r] + bias[n];
    n += 16; if (n < Nc) Crow[n] = acc2[r] + bias[n];
    n += 16; if (n < Nc) Crow[n] = acc3[r] + bias[n];
  }
}

// Coalesced column partial sums: block = 64 cols x row chunk; lane -> column.
__global__ void col_partial_k(float* __restrict__ sums, float* __restrict__ sqs,
                              const float* __restrict__ X, int M, int Nc,
                              int rowsPerBlock) {
  int c = blockIdx.x * 64 + (threadIdx.x & 63);
  int rlane = threadIdx.x >> 6;           // 0..3
  int r0 = blockIdx.y * rowsPerBlock;
  int r1 = r0 + rowsPerBlock; if (r1 > M) r1 = M;
  float s = 0.f, q = 0.f;
  if (c < Nc) {
    for (int r = r0 + rlane; r < r1; r += 4) {
      float v = X[(size_t)r * Nc + c];
      s += v; q += v * v;
    }
  }
  __shared__ float ls[256], lq[256];
  ls[threadIdx.x] = s; lq[threadIdx.x] = q;
  __syncthreads();
  if (rlane == 0 && c < Nc) {
    int t = threadIdx.x;
    float S = ls[t] + ls[t + 64] + ls[t + 128] + ls[t + 192];
    float Q = lq[t] + lq[t + 64] + lq[t + 128] + lq[t + 192];
    atomicAdd(&sums[c], S);
    atomicAdd(&sqs[c], Q);
  }
}

__global__ void stats_finalize_k(float* __restrict__ mean, float* __restrict__ rstd,
                                 const float* __restrict__ sums,
                                 const float* __restrict__ sqs, int M, int Nc) {
  int c = blockIdx.x * blockDim.x + threadIdx.x;
  if (c >= Nc) return;
  float mu = sums[c] / (float)M;
  float var = sqs[c] / (float)M - mu * mu;
  mean[c] = mu;
  rstd[c] = rsqrtf(var + BN_EPS_C);
}

// Y[r,c] = (X[r,c]-mean[c])*rstd[c]*g[c]+be[c] (+relu); row = blockIdx.y
__global__ void bn_apply_k(float* __restrict__ Y, const float* __restrict__ X,
                           const float* __restrict__ mean, const float* __restrict__ rstd,
                           const float* __restrict__ g, const float* __restrict__ be,
                           int Nc, int relu) {
  int c = blockIdx.x * blockDim.x + threadIdx.x;
  if (c >= Nc) return;
  size_t i = (size_t)blockIdx.y * Nc + c;
  float v = (X[i] - mean[c]) * rstd[c] * g[c] + be[c];
  if (relu) v = fmaxf(v, 0.f);
  Y[i] = v;
}

__global__ void fill_zero_k(float* __restrict__ p, int n) {
  for (int i = blockIdx.x * blockDim.x + threadIdx.x; i < n;
       i += gridDim.x * blockDim.x)
    p[i] = 0.f;
}

// hg[batch[n],:] += h[n,:]   (one wave per node)
__global__ void pool_k(float* __restrict__ hg, const float* __restrict__ h,
                       const int* __restrict__ batch, int N) {
  int n = (int)((blockIdx.x * blockDim.x + threadIdx.x) >> 5);
  int lane = threadIdx.x & 31;
  if (n >= N) return;
  int g = batch[n];
  const float* hr = h + (size_t)n * Dg;
  float* gr = hg + (size_t)g * Dg;
  for (int d = lane; d < Dg; d += 32) atomicAdd(gr + d, hr[d]);
}

// Per-row GroupNorm(1,D) fused with head matvec: out[g] = <norm(hg[g]), hw> + hb
__global__ void head_k(float* __restrict__ out, const float* __restrict__ hg,
                       const float* __restrict__ hw, const float* __restrict__ hb) {
  int g = blockIdx.x;
  const float* row = hg + (size_t)g * Dg;
  __shared__ float ss[256], qq[256];
  __shared__ float smu, srstd;
  float s = 0.f, q = 0.f;
  for (int d = threadIdx.x; d < Dg; d += blockDim.x) {
    float v = row[d]; s += v; q += v * v;
  }
  ss[threadIdx.x] = s; qq[threadIdx.x] = q;
  __syncthreads();
  for (int off = 128; off > 0; off >>= 1) {
    if (threadIdx.x < (unsigned)off) {
      ss[threadIdx.x] += ss[threadIdx.x + off];
      qq[threadIdx.x] += qq[threadIdx.x + off];
    }
    __syncthreads();
  }
  if (threadIdx.x == 0) {
    float mu = ss[0] / (float)Dg;
    float var = qq[0] / (float)Dg - mu * mu;
    smu = mu; srstd = rsqrtf(var + BN_EPS_C);
  }
  __syncthreads();
  float mu = smu, rs = srstd;
  float dp = 0.f;
  for (int d = threadIdx.x; d < Dg; d += blockDim.x)
    dp += (row[d] - mu) * rs * hw[d];
  __syncthreads();
  ss[threadIdx.x] = dp;
  __syncthreads();
  for (int off = 128; off > 0; off >>= 1) {
    if (threadIdx.x < (unsigned)off) ss[threadIdx.x] += ss[threadIdx.x + off];
    __syncthreads();
  }
  if (threadIdx.x == 0) out[g] = ss[0] + hb[0];
}

extern "C" void kernel_launch(void* const* d_in, const int* in_sizes, int n_in,
                              void* d_out, int out_size, void* d_ws, size_t ws_size,
                              hipStream_t stream) {
  const int* x          = (const int*)d_in[0];
  const int* edge_index = (const int*)d_in[1];
  const int* edge_attr  = (const int*)d_in[2];
  const int* batch      = (const int*)d_in[3];
  const float* atom_emb = (const float*)d_in[4];
  const float* bond_emb = (const float*)d_in[5];
  const float* eps      = (const float*)d_in[6];
  const float* W1  = (const float*)d_in[7];
  const float* b1  = (const float*)d_in[8];
  const float* g1  = (const float*)d_in[9];
  const float* be1 = (const float*)d_in[10];
  const float* W2  = (const float*)d_in[11];
  const float* b2  = (const float*)d_in[12];
  const float* g2  = (const float*)d_in[13];
  const float* be2 = (const float*)d_in[14];
  const float* head_w = (const float*)d_in[15];
  const float* head_b = (const float*)d_in[16];
  float* out = (float*)d_out;

  const int N = in_sizes[0] / 9;      // 20000
  const int E = in_sizes[2] / 3;      // 320000
  const int* src = edge_index;
  const int* dst = edge_index + E;

  // Workspace layout (floats)
  float* ws   = (float*)d_ws;
  float* h    = ws;                              // N*D
  float* zb   = h    + (size_t)N * Dg;           // N*D  (z, reused as GEMM2 out)
  float* t1   = zb   + (size_t)N * Dg;           // N*H
  float* tab  = t1   + (size_t)N * Hg;           // 512*D
  float* mean = tab  + (size_t)512 * Dg;         // H
  float* rstd = mean + Hg;                       // H
  float* sums = rstd + Hg;                       // H
  float* sqs  = sums + Hg;                       // H
  float* hg   = sqs  + Hg;                       // NG*D
  float* W1p  = hg   + (size_t)NGg * Dg;         // D*HP
  float* W2p  = W1p  + (size_t)Dg * HP;          // H*DP

  const int BT = 256;
  const int gridNodes = (N * 32 + BT - 1) / BT;
  const int gridEdges = (E * 32 + BT - 1) / BT;
  const int Mtiles = (N + 15) / 16;
  const dim3 gemm1Grid(HP / 64, (Mtiles + 7) / 8);
  const dim3 gemm2Grid(DP / 64, (Mtiles + 7) / 8);
  const int gsElem = 2048;
  const int rowChunks = 64;
  const int rowsPerBlock = (N + rowChunks - 1) / rowChunks;
  const dim3 stats1Grid((Hg + 63) / 64, rowChunks);
  const dim3 stats2Grid((Dg + 63) / 64, rowChunks);
  const dim3 bn1Grid((Hg + BT - 1) / BT, N);
  const dim3 bn2Grid((Dg + BT - 1) / BT, N);

  atom_encode_k<<<gridNodes, BT, 0, stream>>>(h, x, atom_emb, N);

  for (int l = 0; l < Lg; ++l) {
    bond_combine_k<<<512, 128, 0, stream>>>(tab, bond_emb + (size_t)l * 24 * Dg);
    z_init_k<<<gsElem, BT, 0, stream>>>(zb, h, eps, l, N * Dg);
    msg_scatter_k<<<gridEdges, BT, 0, stream>>>(zb, h, tab, src, dst, edge_attr, E);

    // z @ W1 + b1 -> t1 ; BN(g1,be1)+ReLU in place
    pad_w_k<<<gsElem, BT, 0, stream>>>(W1p, W1 + (size_t)l * Dg * Hg, Dg, Hg, HP);
    gemm_bias_k<Dg><<<gemm1Grid, BT, 0, stream>>>(t1, zb, W1p,
                                                  b1 + (size_t)l * Hg, N, Hg, HP);
    fill_zero_k<<<8, BT, 0, stream>>>(sums, Hg);
    fill_zero_k<<<8, BT, 0, stream>>>(sqs, Hg);
    col_partial_k<<<stats1Grid, BT, 0, stream>>>(sums, sqs, t1, N, Hg, rowsPerBlock);
    stats_finalize_k<<<(Hg + BT - 1) / BT, BT, 0, stream>>>(mean, rstd, sums, sqs, N, Hg);
    bn_apply_k<<<bn1Grid, BT, 0, stream>>>(t1, t1, mean, rstd,
                                           g1 + (size_t)l * Hg, be1 + (size_t)l * Hg,
                                           Hg, 1);

    // t1 @ W2 + b2 -> zb ; outer BN(g2,be2) (+ReLU except last) -> h
    pad_w_k<<<gsElem, BT, 0, stream>>>(W2p, W2 + (size_t)l * Hg * Dg, Hg, Dg, DP);
    gemm_bias_k<Hg><<<gemm2Grid, BT, 0, stream>>>(zb, t1, W2p,
                                                  b2 + (size_t)l * Dg, N, Dg, DP);
    fill_zero_k<<<8, BT, 0, stream>>>(sums, Dg);
    fill_zero_k<<<8, BT, 0, stream>>>(sqs, Dg);
    col_partial_k<<<stats2Grid, BT, 0, stream>>>(sums, sqs, zb, N, Dg, rowsPerBlock);
    stats_finalize_k<<<(Dg + BT - 1) / BT, BT, 0, stream>>>(mean, rstd, sums, sqs, N, Dg);
    bn_apply_k<<<bn2Grid, BT, 0, stream>>>(h, zb, mean, rstd,
                                           g2 + (size_t)l * Dg, be2 + (size_t)l * Dg,
                                           Dg, (l < Lg - 1) ? 1 : 0);
  }

  fill_zero_k<<<(NGg * Dg + BT - 1) / BT, BT, 0, stream>>>(hg, NGg * Dg);
  pool_k<<<gridNodes, BT, 0, stream>>>(hg, h, batch, N);
  head_k<<<NGg, BT, 0, stream>>>(out, hg, head_w, head_b);
  (void)n_in; (void)out_size; (void)ws_size;
}